// FFTConv_15023795601742
// MI455X (gfx1250) — compile-verified
//
#include <hip/hip_runtime.h>
#include <hip/hip_bf16.h>

typedef __attribute__((ext_vector_type(2))) float v2f;
typedef __attribute__((ext_vector_type(8))) float v8f;
typedef __attribute__((ext_vector_type(4))) unsigned int u32x4;
typedef __attribute__((ext_vector_type(8))) int i32x8;
typedef __attribute__((ext_vector_type(4))) int i32x4;

#define B_SZ  512
#define C_IN  9
#define L_SZ  10000
#define C_MID 64
#define C_OUT 6
#define SEG   100
#define NP    112      // 100 padded to 7*16 (output/M-N dims only; K loops are exactly 100)
#define PITCH 113      // LDS pitch for Ct (odd -> conflict-free)
#define FPITCH 119     // LDS pitch of TDM-padded F: 112 + 7 pads (1 dword per 16)
#define TWO_PI 6.28318530717958647692f

// ---------------- workspace layout (in floats) ----------------
#define OFF_FR   0
#define OFF_FI   (OFF_FR + NP*NP)
#define OFF_TWC  (OFF_FI + NP*NP)
#define OFF_TWS  (OFF_TWC + SEG*SEG)
#define OFF_XSEG (OFF_TWS + SEG*SEG)                  // [i][b][c*100+k]
#define XSEG_SZ  (SEG*B_SZ*C_IN*SEG)
#define OFF_X2   (OFF_XSEG + XSEG_SZ)                 // [b][i][f]

static __device__ __forceinline__ v8f wmma_f32(v2f a, v2f b, v8f c) {
    // V_WMMA_F32_16X16X4_F32 : D = A(16x4) x B(4x16) + C(16x16), wave32
    return __builtin_amdgcn_wmma_f32_16x16x4_f32(
        false, a, false, b, (short)0, c, false, false);
}

// TDM-padded F indexing: element (r,k) of the 112x112 matrix after a
// pad of 1 dword every 16 dwords (rows are 7*16 -> pads align with rows).
static __device__ __forceinline__ int fidx(int r, int k) {
    return r * FPITCH + k + (k >> 4);
}

static __device__ __forceinline__ unsigned lds_off(const void* p) {
    // LDS aperture has zero low-32 base bits: low 32 bits == LDS byte offset.
    return (unsigned)(unsigned long long)p;
}

// Issue one TENSOR_LOAD_TO_LDS. D# built per CDNA5 ISA ch.8:
//   group0: count=1 | lds_addr | global_addr[56:0] | type=2
//   group1: data_size=4B (+optional pad 1dw/16dw), tensor dims/strides, tile dims
//   groups2/3 (+extra words): zero (<=2D tensor)
static __device__ __forceinline__ void tdm_load(const void* gptr, unsigned lds_byte_off,
                                                unsigned td0, unsigned td1,
                                                unsigned tile_w, unsigned tile_h,
                                                unsigned stride0, bool pad16) {
    unsigned long long ga = (unsigned long long)gptr;
    u32x4 g0;
    g0.x = 1u;                                            // count=1, user descriptor
    g0.y = lds_byte_off;
    g0.z = (unsigned)(ga & 0xffffffffull);
    g0.w = (unsigned)((ga >> 32) & 0x01ffffffull) | (2u << 30);  // type=2 ("image")
    i32x8 g1;
    unsigned w0 = (2u << 16);                             // data_size = 4 bytes
    if (pad16) w0 |= (1u << 20) | (3u << 22);             // pad_enable, 1 dword per 16
    g1[0] = (int)w0;
    g1[1] = (int)((td0 & 0xffffu) << 16);                 // tensor_dim0[15:0]
    g1[2] = (int)(((td0 >> 16) & 0xffffu) | ((td1 & 0xffffu) << 16));
    g1[3] = (int)(((td1 >> 16) & 0xffffu) | ((tile_w & 0xffffu) << 16));
    g1[4] = (int)(tile_h & 0xffffu);                      // tile_dim1 (0 = 1D)
    g1[5] = (int)stride0;                                 // tensor_dim0_stride[31:0]
    g1[6] = 0;
    g1[7] = 0;
    i32x4 z4; z4[0] = 0; z4[1] = 0; z4[2] = 0; z4[3] = 0;
    i32x8 z8;
    #pragma unroll
    for (int j = 0; j < 8; j++) z8[j] = 0;
    // clang-23 / therock arity: (g0, g1, g2, g3, extra, cpol)
    __builtin_amdgcn_tensor_load_to_lds(g0, g1, z4, z4, z8, 0);
}

// ---------------- kernel 0: DFT matrix + twiddle tables ----------------
__global__ void k_tables(float* __restrict__ ws) {
    int t = blockIdx.x * blockDim.x + threadIdx.x;
    int stride = gridDim.x * blockDim.x;
    for (int e = t; e < NP*NP; e += stride) {
        int n = e / NP, k = e % NP;
        float cr = 0.f, ci = 0.f;
        if (n < SEG && k < SEG) {
            int m = (n * k) % SEG;
            float ang = -TWO_PI * (float)m / (float)SEG;
            cr = __cosf(ang); ci = __sinf(ang);
        }
        ws[OFF_FR + e] = cr;
        ws[OFF_FI + e] = ci;
    }
    for (int e = t; e < SEG*SEG; e += stride) {
        int n1 = e / SEG, k2 = e % SEG;
        int m = (n1 * k2) % L_SZ;
        float ang = -TWO_PI * (float)m / (float)L_SZ;
        ws[OFF_TWC + e] = __cosf(ang);
        ws[OFF_TWS + e] = __sinf(ang);
    }
}

// ---------------- kernel 1: four-step FFT + log-magnitude (transpose-free) ----------------
// M[n2][n1] = x[n1 + 100*n2] (natural layout, TDM-loaded).
// Bt = F x M           (Bt[k2][n1] == B[n1][k2]; F symmetric)
// Ct = tw .* Bt        (tw[n1*k2] symmetric)
// Ot = Ct x F          (Ot[k2][k1] == X[k2 + 100*k1]) -> segment i=k1, pos=k2
__global__ void k_fft_logmag(const float* __restrict__ x, float* __restrict__ ws) {
    extern __shared__ float lds[];
    float* sM   = lds;                         // [100][100]
    float* sFr  = lds + SEG*SEG;               // padded pitch FPITCH, use fidx()
    float* sFi  = sFr + NP*FPITCH;
    float* sCtr = sFi + NP*FPITCH;             // [112][PITCH]
    float* sCti = sCtr + NP*PITCH;
    const float* twc = ws + OFF_TWC;
    const float* tws = ws + OFF_TWS;
    float* xseg = ws + OFF_XSEG;

    const int bc   = blockIdx.x;               // b*C_IN + c
    const int b    = bc / C_IN;
    const int c    = bc % C_IN;
    const int t    = threadIdx.x;              // 256 threads = 8 waves
    const int lane = t & 31;
    const int wave = t >> 5;
    const int rl   = lane & 15;
    const int hi   = lane >> 4;

    if (wave == 0) {
        tdm_load(x + (size_t)bc * L_SZ, lds_off(sM), L_SZ, 1, L_SZ, 0, L_SZ, false);
        tdm_load(ws + OFF_FR, lds_off(sFr), NP*NP, 1, NP*NP, 0, NP*NP, true);
        tdm_load(ws + OFF_FI, lds_off(sFi), NP*NP, 1, NP*NP, 0, NP*NP, true);
        __builtin_amdgcn_s_wait_tensorcnt(0);
    }
    __syncthreads();

    // ---- Bt = F x M, then Ct = tw .* Bt ----
    for (int tile = wave; tile < 49; tile += 8) {
        int tm = tile / 7, tn = tile % 7;
        v8f btr = {0.f,0.f,0.f,0.f,0.f,0.f,0.f,0.f};
        v8f bti = {0.f,0.f,0.f,0.f,0.f,0.f,0.f,0.f};
        for (int kk = 0; kk < SEG; kk += 4) {  // K = n2, exactly 100
            int ka = kk + 2 * hi;
            int frow = tm*16 + rl;
            v2f afr, afi, bm;
            afr.x = sFr[fidx(frow, ka)];
            afr.y = sFr[fidx(frow, ka + 1)];
            afi.x = sFi[fidx(frow, ka)];
            afi.y = sFi[fidx(frow, ka + 1)];
            int cn = tn*16 + rl;
            bm.x = sM[ ka      * SEG + cn];
            bm.y = sM[(ka + 1) * SEG + cn];
            btr = wmma_f32(afr, bm, btr);
            bti = wmma_f32(afi, bm, bti);
        }
        #pragma unroll
        for (int r = 0; r < 8; r++) {
            int row = tm*16 + r + 8*hi;        // k2
            int col = tn*16 + rl;              // n1
            float cw = 1.f, sw = 0.f;
            if (row < SEG && col < SEG) {
                cw = twc[row*SEG + col];       // tw table is symmetric
                sw = tws[row*SEG + col];
            }
            float brv = btr[r], biv = bti[r];
            sCtr[row*PITCH + col] = brv*cw - biv*sw;
            sCti[row*PITCH + col] = brv*sw + biv*cw;
        }
    }
    __syncthreads();

    // ---- Ot = Ct x F (complex), then 0.5*log(|.|^2) ----
    // f32 WMMA has no A/B negate (NEG = CNeg only): Fi-part subtracted after K loop.
    for (int tile = wave; tile < 49; tile += 8) {
        int tm = tile / 7, tn = tile % 7;
        v8f o1 = {0.f,0.f,0.f,0.f,0.f,0.f,0.f,0.f};   // CtR*Fr
        v8f o2 = {0.f,0.f,0.f,0.f,0.f,0.f,0.f,0.f};   // CtI*Fi
        v8f oi = {0.f,0.f,0.f,0.f,0.f,0.f,0.f,0.f};   // CtR*Fi + CtI*Fr
        for (int kk = 0; kk < SEG; kk += 4) {  // K = n1, exactly 100
            int ka = kk + 2 * hi;
            int crow = tm*16 + rl;
            v2f actr, acti, bfr, bfi;
            actr.x = sCtr[crow*PITCH + ka];
            actr.y = sCtr[crow*PITCH + ka + 1];
            acti.x = sCti[crow*PITCH + ka];
            acti.y = sCti[crow*PITCH + ka + 1];
            int cn = tn*16 + rl;
            bfr.x = sFr[fidx(ka,     cn)];
            bfr.y = sFr[fidx(ka + 1, cn)];
            bfi.x = sFi[fidx(ka,     cn)];
            bfi.y = sFi[fidx(ka + 1, cn)];
            o1 = wmma_f32(actr, bfr, o1);
            o2 = wmma_f32(acti, bfi, o2);
            oi = wmma_f32(actr, bfi, oi);
            oi = wmma_f32(acti, bfr, oi);
        }
        #pragma unroll
        for (int r = 0; r < 8; r++) {
            int row = tm*16 + r + 8*hi;        // k2 = position within segment
            int col = tn*16 + rl;              // k1 = segment index
            if (row < SEG && col < SEG) {
                float orv = o1[r] - o2[r];
                float oiv = oi[r];
                float m2  = orv*orv + oiv*oiv;
                float v   = 0.5f * __logf(m2);
                if (!(__builtin_fabsf(v) < 1e37f)) v = 0.f;   // nan/inf -> 0
                xseg[((size_t)(col*B_SZ + b))*(C_IN*SEG) + c*SEG + row] = v;
            }
        }
    }
}

// ---------------- kernel 2: per-segment linear (einsum1) + sigmoid ----------------
// x2[b][i][f] = sigmoid( sum_{c,k} xseg[i][b][c*100+k] * W1[i][f][c][k] + b1[i][f] )
#define MB 128
__global__ void k_seg_linear(const float* __restrict__ W1, const float* __restrict__ b1,
                             float* __restrict__ ws) {
    extern __shared__ float lds[];
    float* sX = lds;                 // [MB][100]  (TDM 2D tile, pitch 100)
    float* sW = lds + MB*SEG;        // [100][65]  (K x F, manual transpose load)
    const float* xseg = ws + OFF_XSEG;
    float* x2 = ws + OFF_X2;

    const int i    = blockIdx.x;     // segment
    const int mb   = blockIdx.y;     // 0..3 row block
    const int t    = threadIdx.x;
    const int lane = t & 31;
    const int wave = t >> 5;
    const int rl   = lane & 15;
    const int hi   = lane >> 4;
    const int mbase = mb * MB;

    v8f acc[4];
    const v8f z8 = {0.f,0.f,0.f,0.f,0.f,0.f,0.f,0.f};
    #pragma unroll
    for (int nt = 0; nt < 4; nt++) acc[nt] = z8;

    for (int c = 0; c < C_IN; c++) {
        __syncthreads();             // previous chunk fully consumed
        if (wave == 0) {
            // 2D tile: 128 rows x 100 cols out of the [512*100 x 900] xseg plane
            tdm_load(xseg + ((size_t)(i*B_SZ + mbase))*(C_IN*SEG) + c*SEG,
                     lds_off(sX), C_IN*SEG, MB, SEG, MB, C_IN*SEG, false);
        }
        for (int e = t; e < C_MID*SEG; e += 256) {
            int f = e / SEG, k = e % SEG;
            sW[k*65 + f] = W1[(((size_t)i*C_MID + f)*C_IN + c)*SEG + k];
        }
        if (wave == 0) __builtin_amdgcn_s_wait_tensorcnt(0);
        __syncthreads();
        for (int kk = 0; kk < SEG; kk += 4) {
            int ka = kk + 2 * hi;
            v2f a;
            a.x = sX[(wave*16 + rl)*SEG + ka];
            a.y = sX[(wave*16 + rl)*SEG + ka + 1];
            #pragma unroll
            for (int nt = 0; nt < 4; nt++) {
                v2f bb;
                bb.x = sW[ ka      * 65 + nt*16 + rl];
                bb.y = sW[(ka + 1) * 65 + nt*16 + rl];
                acc[nt] = wmma_f32(a, bb, acc[nt]);
            }
        }
    }
    #pragma unroll
    for (int nt = 0; nt < 4; nt++) {
        int f = nt*16 + rl;
        float bias = b1[i*C_MID + f];
        #pragma unroll
        for (int r = 0; r < 8; r++) {
            int row = wave*16 + r + 8*hi;
            float v = acc[nt][r] + bias;
            v = 1.f / (1.f + __expf(-v));
            x2[((size_t)(mbase + row)*SEG + i)*C_MID + f] = v;
        }
    }
}

// ---------------- kernel 3: final contraction (einsum2) ----------------
__global__ void k_final(const float* __restrict__ W2, const float* __restrict__ b2,
                        const float* __restrict__ ws, float* __restrict__ out) {
    __shared__ float red[256][C_OUT];
    const float* x2 = ws + OFF_X2;
    const int b = blockIdx.x, t = threadIdx.x;
    float p[C_OUT] = {0.f, 0.f, 0.f, 0.f, 0.f, 0.f};
    for (int e = t; e < SEG*C_MID; e += 256) {
        int i = e / C_MID, f = e % C_MID;
        float xv = x2[(size_t)b*SEG*C_MID + e];
        #pragma unroll
        for (int o = 0; o < C_OUT; o++)
            p[o] += xv * W2[((size_t)o*C_MID + f)*SEG + i];
    }
    #pragma unroll
    for (int o = 0; o < C_OUT; o++) red[t][o] = p[o];
    __syncthreads();
    for (int s = 128; s > 0; s >>= 1) {
        if (t < s) {
            #pragma unroll
            for (int o = 0; o < C_OUT; o++) red[t][o] += red[t + s][o];
        }
        __syncthreads();
    }
    if (t < C_OUT) out[b*C_OUT + t] = red[0][t] + b2[t];
}

extern "C" void kernel_launch(void* const* d_in, const int* in_sizes, int n_in,
                              void* d_out, int out_size, void* d_ws, size_t ws_size,
                              hipStream_t stream) {
    (void)in_sizes; (void)n_in; (void)out_size; (void)ws_size;
    const float* x  = (const float*)d_in[0];
    const float* W1 = (const float*)d_in[1];
    const float* b1 = (const float*)d_in[2];
    const float* W2 = (const float*)d_in[3];
    const float* b2 = (const float*)d_in[4];
    float* out = (float*)d_out;
    float* ws  = (float*)d_ws;

    k_tables<<<64, 256, 0, stream>>>(ws);

    // sM + 2x padded F + 2x Ct  ->  ~248 KB dynamic LDS (< 320 KB / WGP)
    size_t lds1 = (size_t)(SEG*SEG + 2*NP*FPITCH + 2*NP*PITCH) * sizeof(float);
    k_fft_logmag<<<B_SZ * C_IN, 256, lds1, stream>>>(x, ws);

    dim3 g2(SEG, B_SZ / MB);
    size_t lds2 = (size_t)(MB*SEG + SEG*65) * sizeof(float);   // ~75 KB
    k_seg_linear<<<g2, 256, lds2, stream>>>(W1, b1, ws);

    k_final<<<B_SZ, 256, 0, stream>>>(W2, b2, ws, out);
}